// Attention_40132174414198
// MI455X (gfx1250) — compile-verified
//
#include <hip/hip_runtime.h>
#include <math.h>

// ---------------------------------------------------------------------------
// Tacotron-style location-sensitive attention for MI455X (gfx1250, wave32).
// Bandwidth-bound (~340 MB mandatory traffic -> ~15us floor @ 23.3 TB/s).
// WMMA (V_WMMA_F32_16X16X4_F32, native f32 -> no precision loss) does the
// location-dense GEMM [T x 31] x [31 x 128]. Native V_TANH_F32 epilogue.
// Streaming tensors (memory, processed_memory) use non-temporal loads.
// All reductions are fixed-order (no float atomics) -> deterministic output.
// ---------------------------------------------------------------------------

typedef float v2f __attribute__((ext_vector_type(2)));
typedef float v4f __attribute__((ext_vector_type(4)));
typedef float v8f __attribute__((ext_vector_type(8)));

#if defined(__has_builtin)
#if __has_builtin(__builtin_amdgcn_tanhf)
#define FAST_TANH(x) __builtin_amdgcn_tanhf(x)
#elif __has_builtin(__builtin_amdgcn_tanh_f32)
#define FAST_TANH(x) __builtin_amdgcn_tanh_f32(x)
#else
#define FAST_TANH(x) tanhf(x)
#endif
#else
#define FAST_TANH(x) tanhf(x)
#endif

#define Bsz   64
#define Tlen  2048
#define RNN   1024
#define EMB   512
#define AD    128      // attention dim
#define NF    31       // n_filters
#define KS    31       // conv kernel size
#define PADh  15       // (KS-1)/2
#define TT    128      // t-tile per block in scores kernel
#define NC    8        // t-chunks for context partials (Tlen / 256)

// locf LDS row stride (pad 31->33 to dodge bank conflicts on A fetch)
#define LFS   33
// WdT LDS row stride (pad 128->132 to dodge half-wave bank conflicts)
#define WDS   132
// reduction LDS row stride (pad 16->18: half-wave row sets hit disjoint banks)
#define RDS   18

// ---------------------------------------------------------------------------
// Kernel 1: q[b][d] = sum_k A[b][k] * Wq[d][k]   (b128 loads, LDS-staged A)
// ---------------------------------------------------------------------------
__global__ __launch_bounds__(128)
void k_qproj(const float* __restrict__ A, const float* __restrict__ Wq,
             float* __restrict__ q)
{
    __shared__ float s_a[RNN];
    const int b = blockIdx.x, tid = threadIdx.x;
    {
        const v4f* a4 = (const v4f*)&A[(size_t)b * RNN];
        v4f* s4 = (v4f*)s_a;
        for (int i = tid; i < RNN / 4; i += 128) s4[i] = a4[i];
    }
    __syncthreads();
    const v4f* w4 = (const v4f*)&Wq[(size_t)tid * RNN];
    const v4f* a4 = (const v4f*)s_a;
    float acc = 0.f;
    #pragma unroll 4
    for (int i = 0; i < RNN / 4; ++i) {
        const v4f wv = w4[i], av = a4[i];
        acc += wv.x * av.x + wv.y * av.y + wv.z * av.z + wv.w * av.w;
    }
    q[b * AD + tid] = acc;
}

// ---------------------------------------------------------------------------
// Kernel 2: fused location-conv + dense (WMMA f32) + tanh + score dot.
// Block = 256 threads (8 waves), one b and TT=128 t-values per block.
// Each wave owns 16 t-values (one WMMA M-tile). Deterministic (no atomics).
// ---------------------------------------------------------------------------
__global__ __launch_bounds__(256)
void k_scores(const float* __restrict__ aw,      // [B,2,T]
              const float* __restrict__ pm,      // [B,T,AD]
              const float* __restrict__ q,       // [B,AD]
              const float* __restrict__ Wconv,   // [NF,2,KS]
              const float* __restrict__ Wdense,  // [AD,NF]
              const float* __restrict__ Wscore,  // [AD]
              const float* __restrict__ bscore,  // [1]
              float* __restrict__ scores)        // [B,T]
{
    __shared__ float s_aw[2][TT + KS - 1];   // t tile + halo
    __shared__ float s_wc[NF * 2 * KS];      // conv weights
    __shared__ float s_locf[TT * LFS];       // conv output, K padded to 32
    __shared__ float s_wdT[32 * WDS];        // W_dense^T, K padded to 32
    __shared__ float s_q[AD];
    __shared__ float s_ws[AD];
    __shared__ float s_red[TT * RDS];        // per-(t, d-lane) partial scores

    const int b   = blockIdx.y;
    const int t0  = blockIdx.x * TT;
    const int tid = threadIdx.x;

    // ---- stage inputs in LDS ----
    for (int idx = tid; idx < 2 * (TT + KS - 1); idx += 256) {
        const int c = idx / (TT + KS - 1);
        const int i = idx % (TT + KS - 1);
        const int g = t0 - PADh + i;
        s_aw[c][i] = (g >= 0 && g < Tlen) ? aw[((size_t)b * 2 + c) * Tlen + g] : 0.f;
    }
    for (int idx = tid; idx < NF * 2 * KS; idx += 256) s_wc[idx] = Wconv[idx];
    for (int idx = tid; idx < 32 * AD; idx += 256) {
        const int k = idx / AD, d = idx % AD;
        s_wdT[k * WDS + d] = (k < NF) ? Wdense[d * NF + k] : 0.f;
    }
    for (int i = tid; i < AD; i += 256) {
        s_q[i]  = q[b * AD + i];
        s_ws[i] = Wscore[i];
    }
    __syncthreads();

    // ---- conv: locf[t][f] = sum_{c,k} aw[c][t+k] * Wconv[f][c][k] ----
    for (int idx = tid; idx < TT * NF; idx += 256) {
        const int t = idx / NF, f = idx % NF;
        const float* wf = &s_wc[f * 2 * KS];
        float acc = 0.f;
        #pragma unroll
        for (int k = 0; k < KS; ++k) acc += s_aw[0][t + k] * wf[k];
        #pragma unroll
        for (int k = 0; k < KS; ++k) acc += s_aw[1][t + k] * wf[KS + k];
        s_locf[t * LFS + f] = acc;
    }
    for (int t = tid; t < TT; t += 256) s_locf[t * LFS + NF] = 0.f;  // K=31 pad
    __syncthreads();

    // ---- WMMA dense + fused tanh/score epilogue ----
    const int wave  = tid >> 5;
    const int lane  = tid & 31;
    const int tg    = wave * 16;          // this wave's 16 t rows
    const int mlow  = lane & 15;
    const int khalf = (lane >> 4) << 1;   // lanes 16-31 handle K+2,K+3

    float pacc[8] = {0.f, 0.f, 0.f, 0.f, 0.f, 0.f, 0.f, 0.f};

    #pragma unroll 1
    for (int dt = 0; dt < AD / 16; ++dt) {
        v8f acc = {0.f, 0.f, 0.f, 0.f, 0.f, 0.f, 0.f, 0.f};
        const int n = dt * 16 + mlow;     // this lane's d column
        #pragma unroll
        for (int kk = 0; kk < 8; ++kk) {
            const int kb = kk * 4 + khalf;
            v2f a, bb;
            a.x  = s_locf[(tg + mlow) * LFS + kb];
            a.y  = s_locf[(tg + mlow) * LFS + kb + 1];
            bb.x = s_wdT[kb * WDS + n];
            bb.y = s_wdT[(kb + 1) * WDS + n];
            acc = __builtin_amdgcn_wmma_f32_16x16x4_f32(
                false, a, false, bb, (short)0, acc, false, false);
        }
        // e = tanh(loc + q + pm);  accumulate e * Wscore[d] per t-row
        const float qd  = s_q[n];
        const float wsd = s_ws[n];
        #pragma unroll
        for (int r = 0; r < 8; ++r) {
            const int M = (lane < 16) ? r : (r + 8);
            const int t = t0 + tg + M;
            const float pv =
                __builtin_nontemporal_load(&pm[((size_t)b * Tlen + t) * AD + n]);
            pacc[r] += FAST_TANH(acc[r] + qd + pv) * wsd;
        }
    }

    // one LDS transpose at the end (replaces 256 serialized bpermutes)
    #pragma unroll
    for (int r = 0; r < 8; ++r) {
        const int M = (lane < 16) ? r : (r + 8);
        s_red[(tg + M) * RDS + mlow] = pacc[r];
    }
    __syncthreads();

    const float bs0 = bscore[0];
    for (int i = tid; i < TT; i += 256) {
        float s = bs0;
        #pragma unroll
        for (int j = 0; j < 16; ++j) s += s_red[i * RDS + j];
        scores[(size_t)b * Tlen + t0 + i] = s;
    }
}

// ---------------------------------------------------------------------------
// Kernel 3: masked softmax over T per batch row -> w (written into d_out)
// ---------------------------------------------------------------------------
__global__ __launch_bounds__(256)
void k_softmax(const float* __restrict__ scores,
               const unsigned char* __restrict__ mask,
               float* __restrict__ wout)
{
    __shared__ float red[256];
    const int b = blockIdx.x, tid = threadIdx.x;
    float sv[Tlen / 256];
    float mx = -INFINITY;
    #pragma unroll
    for (int j = 0; j < Tlen / 256; ++j) {
        const int t = tid + j * 256;
        float s = scores[(size_t)b * Tlen + t];
        if (mask[(size_t)b * Tlen + t]) s = -INFINITY;
        sv[j] = s;
        mx = fmaxf(mx, s);
    }
    red[tid] = mx; __syncthreads();
    for (int off = 128; off > 0; off >>= 1) {
        if (tid < off) red[tid] = fmaxf(red[tid], red[tid + off]);
        __syncthreads();
    }
    mx = red[0]; __syncthreads();

    float sum = 0.f;
    #pragma unroll
    for (int j = 0; j < Tlen / 256; ++j) {
        sv[j] = __expf(sv[j] - mx);
        sum += sv[j];
    }
    red[tid] = sum; __syncthreads();
    for (int off = 128; off > 0; off >>= 1) {
        if (tid < off) red[tid] += red[tid + off];
        __syncthreads();
    }
    const float inv = 1.0f / red[0];
    #pragma unroll
    for (int j = 0; j < Tlen / 256; ++j)
        wout[(size_t)b * Tlen + tid + j * 256] = sv[j] * inv;
}

// ---------------------------------------------------------------------------
// Kernel 4a: context partials. Grid (NC, B); block streams a 256-t chunk of
// `memory` with non-temporal b128 loads + global_prefetch_b8, writes
// partial[tc][b][e].
// ---------------------------------------------------------------------------
__global__ __launch_bounds__(128)
void k_context(const float* __restrict__ w, const float* __restrict__ mem,
               float* __restrict__ part)
{
    __shared__ float s_w[256];
    const int b = blockIdx.y, tc = blockIdx.x, tid = threadIdx.x;
    const int tbase = tc * 256;
    for (int i = tid; i < 256; i += 128) s_w[i] = w[(size_t)b * Tlen + tbase + i];
    __syncthreads();

    const v4f* mb = (const v4f*)&mem[((size_t)b * Tlen + tbase) * EMB];
    v4f acc = {0.f, 0.f, 0.f, 0.f};
    #pragma unroll 4
    for (int t = 0; t < 256; ++t) {
        if (t + 16 < 256)
            __builtin_prefetch((const void*)&mb[(size_t)(t + 16) * (EMB / 4) + tid], 0, 0);
        const float wt = s_w[t];
        const v4f   m  = __builtin_nontemporal_load(&mb[(size_t)t * (EMB / 4) + tid]);
        acc += wt * m;
    }
    v4f* p = (v4f*)&part[((size_t)tc * Bsz + b) * EMB];
    p[tid] = acc;
}

// ---------------------------------------------------------------------------
// Kernel 4b: fixed-order reduction of context partials -> ctx[b][e]
// ---------------------------------------------------------------------------
__global__ __launch_bounds__(256)
void k_ctx_reduce(const float* __restrict__ part, float* __restrict__ ctx)
{
    const int i = blockIdx.x * 256 + threadIdx.x;   // 0 .. B*EMB-1
    float s = 0.f;
    #pragma unroll
    for (int c = 0; c < NC; ++c) s += part[(size_t)c * Bsz * EMB + i];
    ctx[i] = s;
}

// ---------------------------------------------------------------------------
extern "C" void kernel_launch(void* const* d_in, const int* in_sizes, int n_in,
                              void* d_out, int out_size, void* d_ws, size_t ws_size,
                              hipStream_t stream)
{
    const float* A    = (const float*)d_in[0];          // [B,RNN]
    const float* mem  = (const float*)d_in[1];          // [B,T,EMB]
    const float* pm   = (const float*)d_in[2];          // [B,T,AD]
    const float* aw   = (const float*)d_in[3];          // [B,2,T]
    const unsigned char* mask = (const unsigned char*)d_in[4]; // [B,T] bool
    const float* Wq   = (const float*)d_in[5];          // [AD,RNN]
    const float* Wc   = (const float*)d_in[6];          // [NF,2,KS]
    const float* Wd   = (const float*)d_in[7];          // [AD,NF]
    const float* Ws   = (const float*)d_in[8];          // [1,AD]
    const float* bs   = (const float*)d_in[9];          // [1]

    float* out    = (float*)d_out;
    float* ctx    = out;                   // [B,EMB]
    float* wout   = out + Bsz * EMB;       // [B,T]
    float* q      = (float*)d_ws;          // [B,AD]
    float* scores = q + Bsz * AD;          // [B,T]
    float* part   = scores + Bsz * Tlen;   // [NC,B,EMB]

    k_qproj     <<<Bsz, 128, 0, stream>>>(A, Wq, q);
    k_scores    <<<dim3(Tlen / TT, Bsz), 256, 0, stream>>>(aw, pm, q, Wc, Wd, Ws, bs, scores);
    k_softmax   <<<Bsz, 256, 0, stream>>>(scores, mask, wout);
    k_context   <<<dim3(NC, Bsz), 128, 0, stream>>>(wout, mem, part);
    k_ctx_reduce<<<(Bsz * EMB + 255) / 256, 256, 0, stream>>>(part, ctx);
}